// ChamferLoss_83442624626996
// MI455X (gfx1250) — compile-verified
//
#include <hip/hip_runtime.h>

typedef float v2f __attribute__((ext_vector_type(2)));
typedef float v8f __attribute__((ext_vector_type(8)));

#define EPSF 1e-6f
#define NPTS 4096
#define BSZ  8
#define WAVES_PER_BLOCK 8   // 256 threads = 8 wave32

// ---------------------------------------------------------------------------
// Prep: pack each point p as float4 (p0, p1, p2, |p|^2) for both x and y,
// and zero the 8-float output accumulator.
// ---------------------------------------------------------------------------
__global__ void chamfer_prep(const float* __restrict__ x,
                             const float* __restrict__ y,
                             float4* __restrict__ Px,
                             float4* __restrict__ Py,
                             float* __restrict__ out) {
  int idx = blockIdx.x * blockDim.x + threadIdx.x;
  if (blockIdx.x == 0 && threadIdx.x < BSZ) out[threadIdx.x] = 0.0f;
  if (idx < BSZ * NPTS) {
    float a0 = x[idx * 3 + 0], a1 = x[idx * 3 + 1], a2 = x[idx * 3 + 2];
    Px[idx] = make_float4(a0, a1, a2, fmaf(a0, a0, fmaf(a1, a1, a2 * a2)));
    float b0 = y[idx * 3 + 0], b1 = y[idx * 3 + 1], b2 = y[idx * 3 + 2];
    Py[idx] = make_float4(b0, b1, b2, fmaf(b0, b0, fmaf(b1, b1, b2 * b2)));
  }
}

// ---------------------------------------------------------------------------
// Main: one wave handles a 16-query-point strip; loops over all reference
// points in 16-wide tiles. Each tile = 1 b64 load + 1 WMMA + 8 v_min.
//   A'[m] = (-2a0, -2a1, -2a2, 1)      (built once per strip)
//   B'[n] = ( x0,   x1,   x2, |x|^2)   (raw b64 load per tile)
//   C[m][n] = |a_m|^2                  (built once per strip)
//   D = A'B' + C = |a_m - x_n|^2
// blockIdx.z selects direction (query=y,ref=x) vs (query=x,ref=y).
// ---------------------------------------------------------------------------
__global__ void chamfer_wmma(const float4* __restrict__ Px,
                             const float4* __restrict__ Py,
                             float* __restrict__ out) {
  const int lane = threadIdx.x & 31;
  const int wave = threadIdx.x >> 5;
  const int half = lane >> 4;   // 0: lanes 0-15 (K=0,1) ; 1: lanes 16-31 (K=2,3)
  const int r    = lane & 15;   // row (A) / col (B) index within tile
  const int b    = blockIdx.y;
  const int dir  = blockIdx.z;

  const float4* PA = (dir == 0) ? Py : Px;   // query points
  const float4* PB = (dir == 0) ? Px : Py;   // reference points
  PA += (size_t)b * NPTS;
  PB += (size_t)b * NPTS;

  const int i0 = (blockIdx.x * WAVES_PER_BLOCK + wave) * 16;

  // --- A fragment: lane holds K={0,1} (half 0) or K={2,3} (half 1) of row i0+r
  const float* pa = (const float*)(PA + (i0 + r)) + half * 2;
  v2f afrag;
  {
    v2f raw = *(const v2f*)pa;                // (a0,a1) or (a2,|a|^2)
    afrag.x = -2.0f * raw.x;
    afrag.y = half ? 1.0f : (-2.0f * raw.y);  // K=3 slot is the constant 1
  }

  // --- C fragment: c[t] = |a_{i0+t+8*half}|^2 broadcast across the 16 lanes
  v8f cfrag;
#pragma unroll
  for (int t = 0; t < 8; ++t)
    cfrag[t] = ((const float*)(PA + (i0 + t + 8 * half)))[3];

  float mn[8];
#pragma unroll
  for (int t = 0; t < 8; ++t) mn[t] = 3.4e38f;

  const float* pb = (const float*)PB + half * 2;
#pragma unroll 4
  for (int j0 = 0; j0 < NPTS; j0 += 16) {
    // B fragment: lane holds K={0,1} or K={2,3} of column j0+r : one b64 load
    v2f bfrag = *(const v2f*)(pb + (size_t)(j0 + r) * 4);
    v8f d = __builtin_amdgcn_wmma_f32_16x16x4_f32(
        false, afrag, false, bfrag, (short)0, cfrag, false, false);
#pragma unroll
    for (int t = 0; t < 8; ++t) mn[t] = fminf(mn[t], d[t]);
  }

  // --- reduce each row's min across its 16 lanes (stays within each half)
  float ssum = 0.0f;
#pragma unroll
  for (int t = 0; t < 8; ++t) {
    float m = mn[t];
    m = fminf(m, __shfl_xor(m, 8, 32));
    m = fminf(m, __shfl_xor(m, 4, 32));
    m = fminf(m, __shfl_xor(m, 2, 32));
    m = fminf(m, __shfl_xor(m, 1, 32));
    ssum += sqrtf(EPSF + fmaxf(m, 0.0f));    // valid on lanes r==0
  }
  if (r == 0)
    atomicAdd(out + b, ssum * (1.0f / (float)NPTS));
}

extern "C" void kernel_launch(void* const* d_in, const int* in_sizes, int n_in,
                              void* d_out, int out_size, void* d_ws, size_t ws_size,
                              hipStream_t stream) {
  const float* x = (const float*)d_in[0];
  const float* y = (const float*)d_in[1];
  float* out = (float*)d_out;

  float4* Px = (float4*)d_ws;                         // 32768 * 16B = 512 KB
  float4* Py = Px + (size_t)BSZ * NPTS;               // + 512 KB

  int npts_total = BSZ * NPTS;
  chamfer_prep<<<(npts_total + 255) / 256, 256, 0, stream>>>(x, y, Px, Py, out);

  dim3 grid(NPTS / 16 / WAVES_PER_BLOCK, BSZ, 2);     // (32, 8, 2)
  chamfer_wmma<<<grid, WAVES_PER_BLOCK * 32, 0, stream>>>(Px, Py, out);
}